// Spline_10591389352571
// MI455X (gfx1250) — compile-verified
//
#include <hip/hip_runtime.h>

typedef __attribute__((ext_vector_type(2))) float v2f;
typedef __attribute__((ext_vector_type(8))) float v8f;

#define DEGP1 4  // cubic: 4 rows of coefficients, 2 dims each -> 8 floats/segment

// ---------------------------------------------------------------------------
// Kernel 1: fold Bezier matrix into per-segment coefficients with WMMA.
//   coeffs[s][j][k] = sum_i BZ[j][i] * pts[s][i][k]
// One wave handles 8 segments via a single V_WMMA_F32_16X16X4_F32:
//   A (16x4)  = BEZIER_M stacked 4x (rows i -> BZ[i&3][*])
//   B (4x16)  = point j of segment (col/2), dim (col&1)
//   D (16x16) rows 0..3 (lanes 0-15, VGPRs 0-3) = coeff rows for 8 segments.
// ---------------------------------------------------------------------------
__global__ void __launch_bounds__(32)
spline_coeff_wmma(const float* __restrict__ ctrl,    // (S*2, 2) flat
                  const float* __restrict__ joint,   // (S+1, 2) flat
                  float* __restrict__ coeffs,        // (S, 4, 2) flat
                  int S)
{
    const int lane  = threadIdx.x;     // 0..31, one wave per block
    const int half  = lane >> 4;       // 0: K=0,1 / B rows 0,1   1: K=2,3 / B rows 2,3
    const int low   = lane & 15;
    const int sBase = blockIdx.x * 8;

    // --- A operand: row (low), BZ row = low & 3 (replicated 4x down the M dim)
    const int r = low & 3;
    float m0, m1, m2, m3;  // BZ[r][0..3]
    if (r == 0)      { m0 =  1.f; m1 =  0.f; m2 =  0.f; m3 = 0.f; }
    else if (r == 1) { m0 = -3.f; m1 =  3.f; m2 =  0.f; m3 = 0.f; }
    else if (r == 2) { m0 =  3.f; m1 = -6.f; m2 =  3.f; m3 = 0.f; }
    else             { m0 = -1.f; m1 =  3.f; m2 = -3.f; m3 = 1.f; }
    v2f a;
    a.x = half ? m2 : m0;
    a.y = half ? m3 : m1;

    // --- B operand: column = low -> segment sBase + low/2, dim = low&1
    const int col = low;
    int seg = sBase + (col >> 1);
    const int segc = (seg < S) ? seg : (S - 1);    // clamp for safe loads
    const int k = col & 1;
    // point rows: P0=joint[s], P1=ctrl[2s], P2=ctrl[2s+1], P3=joint[s+1]
    const float* src0 = half ? (ctrl  + segc * 4 + 2 + k)      // P2
                             : (joint + segc * 2 + k);         // P0
    const float* src1 = half ? (joint + (segc + 1) * 2 + k)    // P3
                             : (ctrl  + segc * 4 + k);         // P1
    v2f b;
    b.x = *src0;
    b.y = *src1;

    v8f c = {};
    // D = A x B + 0  -> v_wmma_f32_16x16x4_f32 (EXEC all-ones here)
    v8f d = __builtin_amdgcn_wmma_f32_16x16x4_f32(
        /*neg_a=*/false, a, /*neg_b=*/false, b,
        /*c_mod=*/(short)0, c, /*reuse_a=*/false, /*reuse_b=*/false);

    // Lanes 0-15 hold rows M=0..3 of the product in d[0..3] (coeff rows 0..3).
    if (half == 0 && seg < S) {
        float* dst = coeffs + seg * 8 + k;  // (S,4,2): row stride = 2 floats
        dst[0] = d[0];
        dst[2] = d[1];
        dst[4] = d[2];
        dst[6] = d[3];
    }
}

// ---------------------------------------------------------------------------
// Per-sample evaluation from precomputed coefficients (Horner, 2 dims).
// ---------------------------------------------------------------------------
__device__ __forceinline__ void eval_coeff(float tt,
                                           const float4* __restrict__ c4,
                                           int S, float Sf,
                                           float& ox, float& oy)
{
    float u  = tt * Sf;
    float fs = floorf(u);
    float lt = u - fs;
    int  seg = (int)fs;
    if (seg >= S) { seg = S - 1; lt = 1.0f; }   // 4096*nextbelow(1) can round to 4096
    if (seg < 0)  { seg = 0; }
    const float4 c01 = c4[seg * 2 + 0];  // {c0x,c0y,c1x,c1y}
    const float4 c23 = c4[seg * 2 + 1];  // {c2x,c2y,c3x,c3y}
    ox = fmaf(fmaf(fmaf(c23.z, lt, c23.x), lt, c01.z), lt, c01.x);
    oy = fmaf(fmaf(fmaf(c23.w, lt, c23.y), lt, c01.w), lt, c01.y);
}

__global__ void __launch_bounds__(256)
spline_eval(const float* __restrict__ t,
            const float* __restrict__ coeffs,
            float* __restrict__ out,
            int N, int S)
{
    const float Sf = (float)S;
    const float4* c4 = reinterpret_cast<const float4*>(coeffs);
    const int idx  = blockIdx.x * blockDim.x + threadIdx.x;
    const int base = idx * 4;

    if (base + 3 < N) {
        // hint t 4096 floats (four blocks) ahead into cache (global_prefetch_b8)
        __builtin_prefetch(t + base + 4096, 0, 1);
        const float4 tv = reinterpret_cast<const float4*>(t)[idx];
        float4 o0, o1;
        eval_coeff(tv.x, c4, S, Sf, o0.x, o0.y);
        eval_coeff(tv.y, c4, S, Sf, o0.z, o0.w);
        eval_coeff(tv.z, c4, S, Sf, o1.x, o1.y);
        eval_coeff(tv.w, c4, S, Sf, o1.z, o1.w);
        float4* o4 = reinterpret_cast<float4*>(out);
        o4[idx * 2 + 0] = o0;
        o4[idx * 2 + 1] = o1;
    } else {
        for (int n = base; n < N; ++n) {
            float x, y;
            eval_coeff(t[n], c4, S, Sf, x, y);
            out[n * 2 + 0] = x;
            out[n * 2 + 1] = y;
        }
    }
}

// ---------------------------------------------------------------------------
// Fallback (workspace too small): fold Bezier matrix inline per sample.
// ---------------------------------------------------------------------------
__device__ __forceinline__ void eval_raw(float tt,
                                         const float* __restrict__ ctrl,
                                         const float* __restrict__ joint,
                                         int S, float Sf,
                                         float& ox, float& oy)
{
    float u  = tt * Sf;
    float fs = floorf(u);
    float lt = u - fs;
    int  seg = (int)fs;
    if (seg >= S) { seg = S - 1; lt = 1.0f; }
    if (seg < 0)  { seg = 0; }
    const float2 P0 = reinterpret_cast<const float2*>(joint)[seg];
    const float2 P3 = reinterpret_cast<const float2*>(joint)[seg + 1];
    const float4 P12 = reinterpret_cast<const float4*>(ctrl)[seg]; // P1.xy, P2.xy
    // power-basis coefficients: c0=P0, c1=3(P1-P0), c2=3P0-6P1+3P2, c3=-P0+3P1-3P2+P3
    float c1x = 3.f * (P12.x - P0.x),            c1y = 3.f * (P12.y - P0.y);
    float c2x = 3.f * P0.x - 6.f * P12.x + 3.f * P12.z;
    float c2y = 3.f * P0.y - 6.f * P12.y + 3.f * P12.w;
    float c3x = -P0.x + 3.f * P12.x - 3.f * P12.z + P3.x;
    float c3y = -P0.y + 3.f * P12.y - 3.f * P12.w + P3.y;
    ox = fmaf(fmaf(fmaf(c3x, lt, c2x), lt, c1x), lt, P0.x);
    oy = fmaf(fmaf(fmaf(c3y, lt, c2y), lt, c1y), lt, P0.y);
}

__global__ void __launch_bounds__(256)
spline_eval_raw(const float* __restrict__ t,
                const float* __restrict__ ctrl,
                const float* __restrict__ joint,
                float* __restrict__ out,
                int N, int S)
{
    const float Sf = (float)S;
    const int idx  = blockIdx.x * blockDim.x + threadIdx.x;
    const int base = idx * 4;
    if (base + 3 < N) {
        const float4 tv = reinterpret_cast<const float4*>(t)[idx];
        float4 o0, o1;
        eval_raw(tv.x, ctrl, joint, S, Sf, o0.x, o0.y);
        eval_raw(tv.y, ctrl, joint, S, Sf, o0.z, o0.w);
        eval_raw(tv.z, ctrl, joint, S, Sf, o1.x, o1.y);
        eval_raw(tv.w, ctrl, joint, S, Sf, o1.z, o1.w);
        float4* o4 = reinterpret_cast<float4*>(out);
        o4[idx * 2 + 0] = o0;
        o4[idx * 2 + 1] = o1;
    } else {
        for (int n = base; n < N; ++n) {
            float x, y;
            eval_raw(t[n], ctrl, joint, S, Sf, x, y);
            out[n * 2 + 0] = x;
            out[n * 2 + 1] = y;
        }
    }
}

extern "C" void kernel_launch(void* const* d_in, const int* in_sizes, int n_in,
                              void* d_out, int out_size, void* d_ws, size_t ws_size,
                              hipStream_t stream)
{
    const float* t     = (const float*)d_in[0];          // (N,)
    const float* ctrl  = (const float*)d_in[1];          // (S*2, 2)
    const float* joint = (const float*)d_in[2];          // (S+1, 2)
    float* out = (float*)d_out;                          // (N, 2)

    const int N = in_sizes[0];
    const int S = in_sizes[2] / 2 - 1;                   // joint_points has S+1 rows

    const int nThreads = (N + 3) / 4;
    const int evalBlocks = (nThreads + 255) / 256;
    const size_t coeffBytes = (size_t)S * 8 * sizeof(float);

    if (ws_size >= coeffBytes) {
        float* coeffs = (float*)d_ws;
        // Phase 1: WMMA coefficient fold (8 segments per wave)
        spline_coeff_wmma<<<(S + 7) / 8, 32, 0, stream>>>(ctrl, joint, coeffs, S);
        // Phase 2: bandwidth-bound evaluation
        spline_eval<<<evalBlocks, 256, 0, stream>>>(t, coeffs, out, N, S);
    } else {
        spline_eval_raw<<<evalBlocks, 256, 0, stream>>>(t, ctrl, joint, out, N, S);
    }
}